// LocalNorm_8057358648424
// MI455X (gfx1250) — compile-verified
//
#include <hip/hip_runtime.h>
#include <hip/hip_bf16.h>
#include <math.h>

typedef __attribute__((ext_vector_type(2))) float v2f;
typedef __attribute__((ext_vector_type(8))) float v8f;

// Local contrast normalization, 33x33 box, zero pad 16, count_include_pad=False.
// PASS 0: src = x,        vertical sums of x,  horizontal via WMMA -> mean; dst = c = x - mean
// PASS 1: src = c,        vertical sums of c^2, horizontal via WMMA -> var;  dst = c/(sqrt(var)+eps)
template <int PASS>
__global__ __launch_bounds__(128) void lcn_pass(const float* __restrict__ src,
                                                float* __restrict__ dst) {
    constexpr int H  = 80;
    constexpr int W  = 3000;
    constexpr int TW = 64;        // output columns per block
    constexpr int LW = TW + 32;   // logical tile width incl. 16+16 halo
    constexpr int LP = LW + 1;    // LDS pitch (97: coprime with 64 banks)

    __shared__ float xs[H * LP];  // tile of src (x in pass0, c in pass1)
    __shared__ float vs[H * LP];  // vertical 33-window sums (of x or c^2)

    const int tile = blockIdx.x;
    const int img  = blockIdx.y;
    const int c0   = tile * TW;               // first output column of this tile
    const int tid  = threadIdx.x;
    const size_t ibase = (size_t)img * H * W;

    // ---- load tile (coalesced along W); zero-fill outside the image ----
    for (int t = tid; t < H * LW; t += 128) {
        const int r  = t / LW;
        const int cl = t - r * LW;
        const int gc = c0 - 16 + cl;
        float v = 0.0f;
        if (gc >= 0 && gc < W) v = src[ibase + (size_t)r * W + gc];
        xs[r * LP + cl] = v;
    }
    __syncthreads();

    // ---- vertical sliding 33-window sum over H (one column per thread) ----
    if (tid < LW) {
        const int cl = tid;
        float s = 0.0f;
        for (int r = 0; r <= 16; ++r) {
            const float v = xs[r * LP + cl];
            s += (PASS == 0) ? v : v * v;
        }
        vs[cl] = s;
        for (int i = 1; i < H; ++i) {
            const int add = i + 16, sub = i - 17;
            if (add < H)  { const float v = xs[add * LP + cl]; s += (PASS == 0) ? v : v * v; }
            if (sub >= 0) { const float v = xs[sub * LP + cl]; s -= (PASS == 0) ? v : v * v; }
            vs[i * LP + cl] = s;
        }
    }
    __syncthreads();

    // ---- horizontal 33-window sum as banded-ones matmul via f32 WMMA ----
    const int lane   = tid & 31;
    const int waveId = tid >> 5;       // 4 waves
    const int lm     = lane & 15;      // N (col) for B/C/D, M (row) for A
    const int half   = lane >> 4;      // lane group selector

    // 5 row-blocks x 4 col-tiles = 20 16x16 output tiles; 5 per wave
    for (int t5 = waveId; t5 < (H / 16) * (TW / 16); t5 += 4) {
        const int rb = (t5 / (TW / 16)) * 16;   // row-block base
        const int ct = (t5 % (TW / 16)) * 16;   // col-tile base (output-local)

        v8f acc0 = {};
        v8f acc1 = {};
        // K window: 48 input columns (LDS cols ct .. ct+47), 12 chunks of 4.
        #pragma unroll
        for (int kc = 0; kc < 12; ++kc) {
            const int kb = kc * 4 + 2 * half;   // K index of .x (ISA A/B VGPR layout)
            v2f a, b;
            // A: 16x4 slab of vertical sums. lanes 0-15 rows M=0..15;
            //    VGPR0 holds K = 2*half, VGPR1 holds K = 2*half+1.
            a.x = vs[(rb + lm) * LP + ct + kb];
            a.y = vs[(rb + lm) * LP + ct + kb + 1];
            // B: banded ones. out col n gets input k iff n <= k <= n+32.
            b.x = (lm <= kb     && kb     <= lm + 32) ? 1.0f : 0.0f;
            b.y = (lm <= kb + 1 && kb + 1 <= lm + 32) ? 1.0f : 0.0f;
            if (kc & 1)
                acc1 = __builtin_amdgcn_wmma_f32_16x16x4_f32(
                    false, a, false, b, (short)0, acc1, false, false);
            else
                acc0 = __builtin_amdgcn_wmma_f32_16x16x4_f32(
                    false, a, false, b, (short)0, acc0, false, false);
        }

        // D layout: VGPR v -> rows rb+v (lanes 0-15) and rb+v+8 (lanes 16-31), col = lm.
        #pragma unroll
        for (int v = 0; v < 8; ++v) {
            const float S  = acc0[v] + acc1[v];
            const int   i  = rb + v + 8 * half;
            const int   gc = c0 + ct + lm;
            if (gc < W) {
                const int ch  = min(i + 16, H - 1) - max(i - 16, 0) + 1;
                const int cw  = min(gc + 16, W - 1) - max(gc - 16, 0) + 1;
                const float cnt  = (float)(ch * cw);
                const float cval = xs[i * LP + ct + lm + 16];
                if (PASS == 0) {
                    const float mean = S / cnt;
                    dst[ibase + (size_t)i * W + gc] = cval - mean;
                } else {
                    const float var = fmaxf(S / cnt, 0.0f);
                    dst[ibase + (size_t)i * W + gc] = cval / (sqrtf(var) + 1e-8f);
                }
            }
        }
    }
}

extern "C" void kernel_launch(void* const* d_in, const int* in_sizes, int n_in,
                              void* d_out, int out_size, void* d_ws, size_t ws_size,
                              hipStream_t stream) {
    (void)in_sizes; (void)n_in; (void)out_size; (void)ws_size;
    const float* x   = (const float*)d_in[0];
    float*       out = (float*)d_out;
    float*       cbuf = (float*)d_ws;   // 64*80*3000*4 = 61.44 MB of scratch for c

    constexpr int W = 3000, TW = 64, NB = 64;
    dim3 grid((W + TW - 1) / TW, NB);   // 47 x 64 blocks
    lcn_pass<0><<<grid, 128, 0, stream>>>(x, cbuf);
    lcn_pass<1><<<grid, 128, 0, stream>>>(cbuf, out);
}